// KPFCNN_26285199851879
// MI455X (gfx1250) — compile-verified
//
#include <hip/hip_runtime.h>
#include <hip/hip_bf16.h>

typedef __attribute__((ext_vector_type(2))) float v2f;
typedef __attribute__((ext_vector_type(8))) float v8f;

#define NBATCH 16
#define GROUPS 64          // partial vectors per batch (multiple of 4 for WMMA K=4 chunks)
#define BLK    256         // 8 waves of 32

// ---------------------------------------------------------------------------
// Kernel 1: fused one-pass weighted reduction.
// Per batch b accumulates 16 values:
//   n=0..8  : M[c][d]  = sum w * src_c * ref_d
//   n=9..11 : s_raw[c] = sum w * src_c
//   n=12..14: r_raw[d] = sum w * ref_d
//   n=15    : S0       = sum w
// Writes partials[b][group][16] to workspace.
// ---------------------------------------------------------------------------
__global__ __launch_bounds__(BLK) void kabsch_reduce(
    const float* __restrict__ src, const float* __restrict__ ref,
    const float* __restrict__ wts, float* __restrict__ partials, int N)
{
    const int b   = blockIdx.y;
    const int tid = blockIdx.x * BLK + threadIdx.x;     // thread id within batch
    const int T   = GROUPS * BLK;                       // threads per batch

    const float* sp = src + (size_t)b * N * 3;
    const float* rp = ref + (size_t)b * N * 3;
    const float* wp = wts + (size_t)b * N;

    float acc[16];
#pragma unroll
    for (int i = 0; i < 16; ++i) acc[i] = 0.0f;

    for (int n = tid; n < N; n += T) {
        float w = wp[n];
        w = (w < 0.0f) ? 0.0f : w;                      // WEIGHT_THRESH = 0
        const float* s3 = sp + n * 3;
        const float* r3 = rp + n * 3;
        float sx = s3[0], sy = s3[1], sz = s3[2];
        float rx = r3[0], ry = r3[1], rz = r3[2];
        float wsx = w * sx, wsy = w * sy, wsz = w * sz;
        acc[0]  += wsx * rx; acc[1]  += wsx * ry; acc[2]  += wsx * rz;
        acc[3]  += wsy * rx; acc[4]  += wsy * ry; acc[5]  += wsy * rz;
        acc[6]  += wsz * rx; acc[7]  += wsz * ry; acc[8]  += wsz * rz;
        acc[9]  += wsx;      acc[10] += wsy;      acc[11] += wsz;
        acc[12] += w * rx;   acc[13] += w * ry;   acc[14] += w * rz;
        acc[15] += w;
    }

    // wave32 shuffle reduction
#pragma unroll
    for (int i = 0; i < 16; ++i) {
#pragma unroll
        for (int off = 16; off > 0; off >>= 1)
            acc[i] += __shfl_down(acc[i], off, 32);
    }

    __shared__ float lds[(BLK / 32) * 16];
    const int lane = threadIdx.x & 31;
    const int wave = threadIdx.x >> 5;
    if (lane == 0) {
#pragma unroll
        for (int i = 0; i < 16; ++i) lds[wave * 16 + i] = acc[i];
    }
    __syncthreads();

    if (threadIdx.x < 16) {
        float t = 0.0f;
#pragma unroll
        for (int w = 0; w < BLK / 32; ++w) t += lds[w * 16 + threadIdx.x];
        partials[((size_t)b * GROUPS + blockIdx.x) * 16 + threadIdx.x] = t;
    }
}

// ---------------------------------------------------------------------------
// Jacobi rotation on symmetric 3x3 A, accumulating right-vectors in V.
// ---------------------------------------------------------------------------
__device__ inline void jacobi_rot(float (&A)[3][3], float (&V)[3][3], int p, int q)
{
    float apq = A[p][q];
    if (fabsf(apq) < 1e-30f) return;
    float theta = (A[q][q] - A[p][p]) / (2.0f * apq);
    float t   = copysignf(1.0f, theta) / (fabsf(theta) + sqrtf(theta * theta + 1.0f));
    float cth = rsqrtf(t * t + 1.0f);
    float sth = t * cth;
    float app = A[p][p], aqq = A[q][q];
    A[p][p] = app - t * apq;
    A[q][q] = aqq + t * apq;
    A[p][q] = 0.0f; A[q][p] = 0.0f;
    int r = 3 - p - q;
    float arp = A[r][p], arq = A[r][q];
    A[r][p] = cth * arp - sth * arq; A[p][r] = A[r][p];
    A[r][q] = sth * arp + cth * arq; A[q][r] = A[r][q];
#pragma unroll
    for (int i = 0; i < 3; ++i) {
        float vip = V[i][p], viq = V[i][q];
        V[i][p] = cth * vip - sth * viq;
        V[i][q] = sth * vip + cth * viq;
    }
}

__device__ inline float det3c(const float (&u)[3][3])
{   // determinant, columns are vectors: u[row][col]
    return u[0][0] * (u[1][1] * u[2][2] - u[1][2] * u[2][1])
         - u[0][1] * (u[1][0] * u[2][2] - u[1][2] * u[2][0])
         + u[0][2] * (u[1][0] * u[2][1] - u[1][1] * u[2][0]);
}

// ---------------------------------------------------------------------------
// Kernel 2: one block, 16 waves; wave b combines GROUPS partial vectors of
// batch b with chained V_WMMA_F32_16X16X4_F32 (A = ones, B = 4 partials per
// step, C accumulates column sums), then lane 0 solves the 3x3 Kabsch SVD.
// ---------------------------------------------------------------------------
__global__ __launch_bounds__(NBATCH * 32) void kabsch_solve(
    const float* __restrict__ partials, float* __restrict__ out)
{
    const int lane = threadIdx.x & 31;
    const int b    = threadIdx.x >> 5;                  // batch = wave id

    const float* pb = partials + (size_t)b * GROUPS * 16;
    const int half = lane >> 4;                         // 0: lanes 0-15, 1: lanes 16-31
    const int col  = lane & 15;

    v2f av; av[0] = 1.0f; av[1] = 1.0f;                 // A = all-ones 16x4
    v8f c = {};

    // EXEC is all-ones throughout this loop (no divergence before it ends).
    for (int g0 = 0; g0 < GROUPS; g0 += 4) {
        v2f bv;
        bv[0] = pb[(g0 + 2 * half + 0) * 16 + col];
        bv[1] = pb[(g0 + 2 * half + 1) * 16 + col];
        c = __builtin_amdgcn_wmma_f32_16x16x4_f32(
                /*neg_a=*/false, av, /*neg_b=*/false, bv,
                /*c_mod=*/(short)0, c, /*reuse_a=*/false, /*reuse_b=*/false);
    }

    // Row 0 of C lives in c[0] on lanes 0..15 (every row holds the same sums).
    float c0 = c[0];
    float tot[16];
#pragma unroll
    for (int n = 0; n < 16; ++n) tot[n] = __shfl(c0, n, 32);

    if (lane == 0) {
        // Recover H = M/D - (2 - Sw) * sbar * rbar^T   (single-pass identity)
        float D   = tot[15] + 1e-5f;
        float inv = 1.0f / D;
        float Sw  = tot[15] * inv;
        float sb[3], rb[3];
#pragma unroll
        for (int i = 0; i < 3; ++i) { sb[i] = tot[9 + i] * inv; rb[i] = tot[12 + i] * inv; }
        float k2 = 2.0f - Sw;
        float h[3][3];
#pragma unroll
        for (int i = 0; i < 3; ++i)
#pragma unroll
            for (int j = 0; j < 3; ++j)
                h[i][j] = tot[i * 3 + j] * inv - k2 * sb[i] * rb[j];

        // A = H^T H (symmetric), Jacobi eigendecomposition -> V, eigenvalues
        float A[3][3];
#pragma unroll
        for (int i = 0; i < 3; ++i)
#pragma unroll
            for (int j = 0; j < 3; ++j)
                A[i][j] = h[0][i] * h[0][j] + h[1][i] * h[1][j] + h[2][i] * h[2][j];

        float V[3][3] = {{1, 0, 0}, {0, 1, 0}, {0, 0, 1}};
        for (int sweep = 0; sweep < 12; ++sweep) {
            jacobi_rot(A, V, 0, 1);
            jacobi_rot(A, V, 0, 2);
            jacobi_rot(A, V, 1, 2);
        }

        // sort eigenpairs descending (columns of V)
        float e[3] = {A[0][0], A[1][1], A[2][2]};
#pragma unroll
        for (int i = 0; i < 2; ++i)
#pragma unroll
            for (int j = i + 1; j < 3; ++j)
                if (e[j] > e[i]) {
                    float te = e[i]; e[i] = e[j]; e[j] = te;
#pragma unroll
                    for (int r = 0; r < 3; ++r) {
                        float tv = V[r][i]; V[r][i] = V[r][j]; V[r][j] = tv;
                    }
                }

        // U columns: u_i = H v_i / sigma_i
        float U[3][3];
#pragma unroll
        for (int i = 0; i < 3; ++i) {
            float sig = sqrtf(fmaxf(e[i], 0.0f));
            float is  = 1.0f / fmaxf(sig, 1e-20f);
#pragma unroll
            for (int r = 0; r < 3; ++r)
                U[r][i] = (h[r][0] * V[0][i] + h[r][1] * V[1][i] + h[r][2] * V[2][i]) * is;
        }

        float sgn = (det3c(U) * det3c(V) < 0.0f) ? -1.0f : 1.0f;
        float d[3] = {1.0f, 1.0f, sgn};

        // R = V diag(d) U^T ;  t = rbar - R sbar
        float R[3][3];
#pragma unroll
        for (int i = 0; i < 3; ++i)
#pragma unroll
            for (int j = 0; j < 3; ++j)
                R[i][j] = V[i][0] * d[0] * U[j][0] + V[i][1] * d[1] * U[j][1]
                        + V[i][2] * d[2] * U[j][2];

#pragma unroll
        for (int i = 0; i < 3; ++i) {
#pragma unroll
            for (int j = 0; j < 3; ++j) out[b * 9 + i * 3 + j] = R[i][j];
            out[NBATCH * 9 + b * 3 + i] =
                rb[i] - (R[i][0] * sb[0] + R[i][1] * sb[1] + R[i][2] * sb[2]);
        }
    }
}

extern "C" void kernel_launch(void* const* d_in, const int* in_sizes, int n_in,
                              void* d_out, int out_size, void* d_ws, size_t ws_size,
                              hipStream_t stream)
{
    const float* src = (const float*)d_in[0];   // (16, N, 3) f32
    const float* ref = (const float*)d_in[1];   // (16, N, 3) f32
    const float* wts = (const float*)d_in[2];   // (16, N)    f32
    float* out       = (float*)d_out;           // 144 (R) + 48 (t) f32
    float* partials  = (float*)d_ws;            // 16 * GROUPS * 16 floats = 64 KB

    const int N = in_sizes[2] / NBATCH;

    dim3 grid(GROUPS, NBATCH);
    kabsch_reduce<<<grid, BLK, 0, stream>>>(src, ref, wts, partials, N);
    kabsch_solve<<<1, NBATCH * 32, 0, stream>>>(partials, out);
}